// EE_83468394431314
// MI455X (gfx1250) — compile-verified
//
#include <hip/hip_runtime.h>
#include <hip/hip_bf16.h>

typedef float v2f __attribute__((ext_vector_type(2)));
typedef float v8f __attribute__((ext_vector_type(8)));

#define D 128          // feature dim (fixed by the model)
#define NEG_SLOPE 0.2f

// ---------------------------------------------------------------------------
// zero-fill
// ---------------------------------------------------------------------------
__global__ void fill_zero_kernel(float* __restrict__ p, long n) {
    long i = (long)blockIdx.x * blockDim.x + threadIdx.x;
    long stride = (long)gridDim.x * blockDim.x;
    for (; i < n; i += stride) p[i] = 0.0f;
}

// ---------------------------------------------------------------------------
// Edge aggregation: one wave32 per edge.
//   agg[dst] += src_h[src] * ew   (128 floats, float4 per lane, f32 atomics)
//   deg[dst] += 1
// ---------------------------------------------------------------------------
__global__ void edge_agg_kernel(const float* __restrict__ srcH,
                                const int* __restrict__ sidx,
                                const int* __restrict__ didx,
                                const float* __restrict__ ew,
                                float* __restrict__ agg,
                                float* __restrict__ deg,
                                int E) {
    int gtid = blockIdx.x * blockDim.x + threadIdx.x;
    int edge = gtid >> 5;
    int lane = threadIdx.x & 31;
    if (edge >= E) return;

    int s = sidx[edge];
    int d = didx[edge];
    float w = ew ? ew[edge] : 1.0f;

    const float4* srow = (const float4*)(srcH + (size_t)s * D);
    float4 v = srow[lane];                       // lane covers 4 consecutive floats

    float* arow = agg + (size_t)d * D + lane * 4;
    atomicAdd(arow + 0, v.x * w);
    atomicAdd(arow + 1, v.y * w);
    atomicAdd(arow + 2, v.z * w);
    atomicAdd(arow + 3, v.w * w);
    if (lane == 0) atomicAdd(deg + d, 1.0f);
}

// ---------------------------------------------------------------------------
// Per-half K=128 accumulate. Inlined with concrete kernel-arg pointers so the
// backend keeps the GLOBAL address space (global_load_b64, not flat_load).
// W column base pointers are hoisted out of the K loop (constant-index array
// -> fully register-promoted), and the 8 B fragments of each K-step are
// loaded as a batch before the 8 WMMAs so the load waits overlap.
// ---------------------------------------------------------------------------
__device__ __forceinline__ void gemm_accum_half(const float* __restrict__ A,
                                                const float* __restrict__ W, int ldw,
                                                int rA, int c15, int hk,
                                                float rd, v8f acc[8]) {
    const float* ap = A + (size_t)rA * D + hk;
    const float* wp[8];
#pragma unroll
    for (int nt = 0; nt < 8; ++nt)
        wp[nt] = W + (size_t)(nt * 16 + c15) * ldw + hk;

#pragma unroll 4
    for (int k = 0; k < D; k += 4) {
        v2f a = *(const v2f*)(ap + k);
        a *= rd;
        v2f b[8];
#pragma unroll
        for (int nt = 0; nt < 8; ++nt)
            b[nt] = *(const v2f*)(wp[nt] + k);
#pragma unroll
        for (int nt = 0; nt < 8; ++nt)
            acc[nt] = __builtin_amdgcn_wmma_f32_16x16x4_f32(
                false, a, false, b[nt], (short)0, acc[nt], false, false);
    }
}

// ---------------------------------------------------------------------------
// Fused dual-operand GEMM using V_WMMA_F32_16X16X4_F32 (fp32 matrix core):
//   Y[N,128] = A0[N,128] @ W0slice^T
//            + (A1[N,128] * (deg? 1/max(deg,1) : 1)) @ W1slice^T + bias
//   act: 0 = none, 1 = relu, 2 = leaky(0.2)
// One wave per 16-row strip; 8 accumulators cover all 128 output columns.
// ---------------------------------------------------------------------------
__global__ __launch_bounds__(256)
void sage_gemm_kernel(const float* __restrict__ A0,
                      const float* __restrict__ W0, int ldw0,
                      const float* __restrict__ A1,
                      const float* __restrict__ W1, int ldw1,
                      const float* __restrict__ deg,
                      const float* __restrict__ bias,
                      float* __restrict__ Y, int N, int act) {
    int lane = threadIdx.x & 31;
    int wid  = threadIdx.x >> 5;
    int m0 = (blockIdx.x * 8 + wid) * 16;
    if (m0 >= N) return;                  // whole-wave exit; EXEC stays all-ones

    int c15 = lane & 15;                  // column within a 16-wide tile
    int hk  = (lane >> 4) * 2;            // K sub-offset per WMMA f32 A/B layout
    int rA  = m0 + c15;                   // A-fragment row for this lane
    if (rA > N - 1) rA = N - 1;           // clamp (N%16==0 in practice)

    v8f acc[8];
#pragma unroll
    for (int nt = 0; nt < 8; ++nt) acc[nt] = (v8f)0.0f;

    gemm_accum_half(A0, W0, ldw0, rA, c15, hk, 1.0f, acc);
    if (A1 != nullptr) {
        float rd = deg ? (1.0f / fmaxf(deg[rA], 1.0f)) : 1.0f;   // fused mean
        gemm_accum_half(A1, W1, ldw1, rA, c15, hk, rd, acc);
    }

    // epilogue: bias + activation + store (C/D layout: vgpr g -> row m0+g(+8))
    int rbase = m0 + ((lane >> 4) << 3);
#pragma unroll
    for (int nt = 0; nt < 8; ++nt) {
        int col = nt * 16 + c15;
        float bb = bias ? bias[col] : 0.0f;
#pragma unroll
        for (int g = 0; g < 8; ++g) {
            float v = acc[nt][g] + bb;
            if (act == 1)      v = fmaxf(v, 0.0f);
            else if (act == 2) v = (v >= 0.0f) ? v : NEG_SLOPE * v;
            int row = rbase + g;
            if (row < N) Y[(size_t)row * D + col] = v;
        }
    }
}

// ---------------------------------------------------------------------------
// host-side launch helpers
// ---------------------------------------------------------------------------
static void run_gemm(const float* A0, const float* W0, int ldw0,
                     const float* A1, const float* W1, int ldw1,
                     const float* deg, const float* bias,
                     float* Y, int N, int act, hipStream_t s) {
    dim3 blk(256), grd((N + 127) / 128);
    sage_gemm_kernel<<<grd, blk, 0, s>>>(A0, W0, ldw0, A1, W1, ldw1, deg, bias, Y, N, act);
}

static void run_zero(float* p, long n, hipStream_t s) {
    long blocks = (n + 255) / 256;
    if (blocks > 65535) blocks = 65535;
    fill_zero_kernel<<<dim3((unsigned)blocks), dim3(256), 0, s>>>(p, n);
}

static void run_edge(const float* srcH, const int* si, const int* di, const float* ew,
                     float* agg, float* deg, int E, hipStream_t s) {
    dim3 blk(256), grd((E + 7) / 8);   // 8 waves per block, 1 wave per edge
    edge_agg_kernel<<<grd, blk, 0, s>>>(srcH, si, di, ew, agg, deg, E);
}

extern "C" void kernel_launch(void* const* d_in, const int* in_sizes, int n_in,
                              void* d_out, int out_size, void* d_ws, size_t ws_size,
                              hipStream_t stream) {
    const float* ii       = (const float*)d_in[0];
    const float* e_emb    = (const float*)d_in[1];
    const float* k_emb    = (const float*)d_in[2];
    const float* cf_ew    = (const float*)d_in[3];
    const int*   b_src    = (const int*)d_in[4];
    const int*   b_dst    = (const int*)d_in[5];
    const int*   c_src    = (const int*)d_in[6];
    const int*   c_dst    = (const int*)d_in[7];
    const float* ek_self  = (const float*)d_in[8];
    const float* ek_neigh = (const float*)d_in[9];
    const float* ek_bias  = (const float*)d_in[10];
    const float* ee_self  = (const float*)d_in[11];
    const float* ee_neigh = (const float*)d_in[12];
    const float* ee_bias  = (const float*)d_in[13];
    const float* w1_W = (const float*)d_in[14];
    const float* w1_b = (const float*)d_in[15];
    const float* w2_W = (const float*)d_in[16];
    const float* w2_b = (const float*)d_in[17];
    const float* w3_W = (const float*)d_in[18];
    const float* w3_b = (const float*)d_in[19];
    const float* w4_W = (const float*)d_in[20];
    const float* w4_b = (const float*)d_in[21];
    const float* comb_W = (const float*)d_in[22];
    const float* comb_b = (const float*)d_in[23];

    const int n_k = in_sizes[0] / D;
    const int n_e = in_sizes[1] / D;
    const int E   = in_sizes[3];
    const int L   = in_sizes[8] / (D * D);

    // workspace layout (floats): ii2 | agg | deg | t0 | t1 | h_iI
    float* ws   = (float*)d_ws;
    float* ii2  = ws;
    float* agg  = ii2 + (size_t)n_k * D;
    float* deg  = agg + (size_t)n_e * D;                 // contiguous after agg
    size_t degPad = ((size_t)n_e + 3) & ~(size_t)3;      // keep 16B alignment
    float* t0   = deg + degPad;
    float* t1   = t0 + (size_t)n_e * D;
    float* h_iI = t1 + (size_t)n_e * D;
    (void)ws_size; (void)n_in; (void)out_size;

    // 1. ii2 = [ii | k_emb] @ w4^T + w4_b
    run_gemm(ii, w4_W, 2 * D, k_emb, w4_W + D, 2 * D, nullptr, w4_b, ii2, n_k, 0, stream);

    // 2. belong aggregation (src features constant across layers -> aggregate ONCE)
    run_zero(agg, (long)n_e * D + n_e, stream);          // agg + deg in one pass
    run_edge(ii2, b_src, b_dst, nullptr, agg, deg, E, stream);

    // 3. knowledge->employee SAGE stack
    const float* dcur = e_emb;
    float* ping = t0; float* pong = t1;
    for (int i = 0; i < L - 1; ++i) {
        run_gemm(dcur, ek_self + (size_t)i * D * D, D,
                 agg,  ek_neigh + (size_t)i * D * D, D,
                 deg, ek_bias + (size_t)i * D, ping, n_e, /*relu*/1, stream);
        dcur = ping; float* tmp = ping; ping = pong; pong = tmp;
    }
    run_gemm(dcur, ek_self + (size_t)(L - 1) * D * D, D,
             agg,  ek_neigh + (size_t)(L - 1) * D * D, D,
             deg, ek_bias + (size_t)(L - 1) * D, ping, n_e, 0, stream);  // h_iI_pre
    run_gemm(ping, w2_W, D, nullptr, nullptr, 0, nullptr, w2_b, h_iI, n_e, 0, stream);

    // 4. src_com = [h_iI | e_emb] @ w1^T + w1_b   (-> t0, free now)
    run_gemm(h_iI, w1_W, 2 * D, e_emb, w1_W + D, 2 * D, nullptr, w1_b, t0, n_e, 0, stream);

    // 5. collaborate aggregation (edge-weighted), again computed once
    run_zero(agg, (long)n_e * D + n_e, stream);
    run_edge(t0, c_src, c_dst, cf_ew, agg, deg, E, stream);

    // 6. employee->employee SAGE stack
    dcur = e_emb; ping = t1; pong = t0;
    for (int i = 0; i < L - 1; ++i) {
        run_gemm(dcur, ee_self + (size_t)i * D * D, D,
                 agg,  ee_neigh + (size_t)i * D * D, D,
                 deg, ee_bias + (size_t)i * D, ping, n_e, /*relu*/1, stream);
        dcur = ping; float* tmp = ping; ping = pong; pong = tmp;
    }
    run_gemm(dcur, ee_self + (size_t)(L - 1) * D * D, D,
             agg,  ee_neigh + (size_t)(L - 1) * D * D, D,
             deg, ee_bias + (size_t)(L - 1) * D, ping, n_e, 0, stream);  // h_iS_pre
    run_gemm(ping, w3_W, D, nullptr, nullptr, 0, nullptr, w3_b, pong, n_e, 0, stream); // h_iS

    // 7. out = leaky([h_iI | h_iS] @ comb^T + comb_b)
    run_gemm(h_iI, comb_W, 2 * D, pong, comb_W + D, 2 * D,
             nullptr, comb_b, (float*)d_out, n_e, /*leaky*/2, stream);
}